// PointNet2FPModule_25434796327489
// MI455X (gfx1250) — compile-verified
//
#include <hip/hip_runtime.h>

#define NB      16
#define NI_PER  256
#define NS_PER  1024
#define C_IN    256
#define C_SKIP  128
#define NS_TOT  (NB * NS_PER)   // 16384
#define DCAT    387
#define DPAD    388
#define H1S     260             // padded LDS stride for h1 (260 % 64 == 4)

typedef float v2f __attribute__((ext_vector_type(2)));
typedef float v8f __attribute__((ext_vector_type(8)));

// ---------------------------------------------------------------------------
// Kernel 1: 3-NN inverse-distance interpolation + concat -> Hcat[16384, 388]
// One block = 256 fine points, all in the same batch.
// ---------------------------------------------------------------------------
__global__ void __launch_bounds__(256) knn_interp_concat(
    const float* __restrict__ in_x, const float* __restrict__ in_pos,
    const float* __restrict__ skip_x, const float* __restrict__ skip_pos,
    float* __restrict__ Hcat)
{
    __shared__ float sp[NI_PER * 3];
    __shared__ int   sidx[3][256];
    __shared__ float sw[3][256];

    const int tid   = threadIdx.x;
    const int s0    = blockIdx.x * 256;
    const int batch = s0 / NS_PER;

    const float* ip = in_pos + (size_t)batch * NI_PER * 3;
    for (int i = tid; i < NI_PER * 3; i += 256) sp[i] = ip[i];
    __syncthreads();

    const int s = s0 + tid;
    const float px = skip_pos[s*3+0], py = skip_pos[s*3+1], pz = skip_pos[s*3+2];

    float d0 = 3.4e38f, d1 = 3.4e38f, d2 = 3.4e38f;
    int   i0 = 0, i1 = 0, i2 = 0;
    for (int j = 0; j < NI_PER; ++j) {
        float dx = sp[3*j+0] - px;
        float dy = sp[3*j+1] - py;
        float dz = sp[3*j+2] - pz;
        float d  = fmaf(dx, dx, fmaf(dy, dy, dz * dz));
        if (d < d0)      { d2 = d1; i2 = i1; d1 = d0; i1 = i0; d0 = d; i0 = j; }
        else if (d < d1) { d2 = d1; i2 = i1; d1 = d;  i1 = j; }
        else if (d < d2) { d2 = d;  i2 = j; }
    }
    float w0 = 1.f / fmaxf(sqrtf(d0), 1e-10f);
    float w1 = 1.f / fmaxf(sqrtf(d1), 1e-10f);
    float w2 = 1.f / fmaxf(sqrtf(d2), 1e-10f);
    float inv = 1.f / (w0 + w1 + w2 + 1e-16f);
    sw[0][tid] = w0 * inv;  sw[1][tid] = w1 * inv;  sw[2][tid] = w2 * inv;
    sidx[0][tid] = i0;      sidx[1][tid] = i1;      sidx[2][tid] = i2;
    __syncthreads();

    // Cooperative, channel-major pass: thread == channel -> coalesced.
    const float* inx = in_x + (size_t)batch * NI_PER * C_IN;
    const int c = tid;
    for (int p = 0; p < 256; ++p) {
        const int row = s0 + p;
        float v = sw[0][p] * inx[sidx[0][p]*C_IN + c]
                + sw[1][p] * inx[sidx[1][p]*C_IN + c]
                + sw[2][p] * inx[sidx[2][p]*C_IN + c];
        Hcat[(size_t)row * DPAD + c] = v;
        if (c < C_SKIP) {
            Hcat[(size_t)row * DPAD + C_IN + c] = skip_x[(size_t)row * C_SKIP + c];
        } else if (c < C_SKIP + 3) {
            Hcat[(size_t)row * DPAD + C_IN + C_SKIP + (c - C_SKIP)] =
                skip_pos[(size_t)row * 3 + (c - C_SKIP)];
        } else if (c == C_SKIP + 3) {
            Hcat[(size_t)row * DPAD + DCAT] = 0.f;   // K-pad column
        }
    }
}

// ---------------------------------------------------------------------------
// Kernel 2: W1 [387,256] -> W1p [388,256] with a zero row (no OOB reads).
// ---------------------------------------------------------------------------
__global__ void __launch_bounds__(256) pad_w1(const float* __restrict__ W1,
                                              float* __restrict__ W1p)
{
    const int i = blockIdx.x * 256 + threadIdx.x;   // grid covers 388*256 exactly
    W1p[i] = (i < DCAT * 256) ? W1[i] : 0.f;
}

// ---------------------------------------------------------------------------
// Kernel 3: fused  relu(relu(Hcat@W1p + b1) @ W2 + b2)  via f32 WMMA.
// Block = 64 rows x full N=256; 8 waves; wave w owns N-tiles {2w, 2w+1}
// and all 4 M-tiles. h1 lives only in LDS.
// ---------------------------------------------------------------------------
__global__ void __launch_bounds__(256) fused_mlp(
    const float* __restrict__ Hcat, const float* __restrict__ W1p,
    const float* __restrict__ b1,   const float* __restrict__ W2,
    const float* __restrict__ b2,   float* __restrict__ out)
{
    extern __shared__ float lds[];
    float* As  = lds;               // 64 x 388
    float* h1s = lds + 64 * DPAD;   // 64 x 260 (padded stride)

    const int tid   = threadIdx.x;
    const int lane  = tid & 31;
    const int wave  = tid >> 5;
    const int mrow  = lane & 15;           // M (or N) index within a 16-wide tile
    const int khalf = (lane >> 4) << 1;    // 0 or 2 (A/B fragment K split)
    const int mhi   = (lane >> 4) << 3;    // 0 or 8 (C/D row split)
    const int r0    = blockIdx.x * 64;
    const int n0    = wave * 32;

    // Stage A tile: rows are contiguous (stride 388 in both global and LDS).
    {
        const float4* src = (const float4*)(Hcat + (size_t)r0 * DPAD);
        float4*       dst = (float4*)As;
        for (int i = tid; i < (64 * DPAD) / 4; i += 256) dst[i] = src[i];
    }
    __syncthreads();

    // ---- Layer 1: [64,388] x [388,256] ----
    v8f acc[4][2] = {};
    for (int k = 0; k < DPAD; k += 4) {
        v2f bf0, bf1;
        {
            const float* w = W1p + (size_t)(k + khalf) * 256;
            bf0.x = w[n0 + mrow];        bf0.y = w[256 + n0 + mrow];
            bf1.x = w[n0 + 16 + mrow];   bf1.y = w[256 + n0 + 16 + mrow];
        }
#pragma unroll
        for (int mt = 0; mt < 4; ++mt) {
            v2f af = *(const v2f*)(As + (mt * 16 + mrow) * DPAD + k + khalf);
            acc[mt][0] = __builtin_amdgcn_wmma_f32_16x16x4_f32(
                false, af, false, bf0, (short)0, acc[mt][0], false, false);
            acc[mt][1] = __builtin_amdgcn_wmma_f32_16x16x4_f32(
                false, af, false, bf1, (short)0, acc[mt][1], false, false);
        }
    }

    // bias + relu -> h1 in LDS
    {
        const float bb0 = b1[n0 + mrow];
        const float bb1 = b1[n0 + 16 + mrow];
#pragma unroll
        for (int mt = 0; mt < 4; ++mt)
#pragma unroll
            for (int v = 0; v < 8; ++v) {
                const int m = mt * 16 + v + mhi;
                h1s[m * H1S + n0 + mrow]      = fmaxf(acc[mt][0][v] + bb0, 0.f);
                h1s[m * H1S + n0 + 16 + mrow] = fmaxf(acc[mt][1][v] + bb1, 0.f);
            }
    }
    __syncthreads();

    // ---- Layer 2: [64,256] x [256,256] ----
    v8f acc2[4][2] = {};
    for (int k = 0; k < 256; k += 4) {
        v2f bf0, bf1;
        {
            const float* w = W2 + (size_t)(k + khalf) * 256;
            bf0.x = w[n0 + mrow];        bf0.y = w[256 + n0 + mrow];
            bf1.x = w[n0 + 16 + mrow];   bf1.y = w[256 + n0 + 16 + mrow];
        }
#pragma unroll
        for (int mt = 0; mt < 4; ++mt) {
            v2f af = *(const v2f*)(h1s + (mt * 16 + mrow) * H1S + k + khalf);
            acc2[mt][0] = __builtin_amdgcn_wmma_f32_16x16x4_f32(
                false, af, false, bf0, (short)0, acc2[mt][0], false, false);
            acc2[mt][1] = __builtin_amdgcn_wmma_f32_16x16x4_f32(
                false, af, false, bf1, (short)0, acc2[mt][1], false, false);
        }
    }

    // bias + relu -> global output
    {
        const float bb0 = b2[n0 + mrow];
        const float bb1 = b2[n0 + 16 + mrow];
#pragma unroll
        for (int mt = 0; mt < 4; ++mt)
#pragma unroll
            for (int v = 0; v < 8; ++v) {
                const int m = r0 + mt * 16 + v + mhi;
                out[(size_t)m * 256 + n0 + mrow]      = fmaxf(acc2[mt][0][v] + bb0, 0.f);
                out[(size_t)m * 256 + n0 + 16 + mrow] = fmaxf(acc2[mt][1][v] + bb1, 0.f);
            }
    }
}

// ---------------------------------------------------------------------------
extern "C" void kernel_launch(void* const* d_in, const int* in_sizes, int n_in,
                              void* d_out, int out_size, void* d_ws, size_t ws_size,
                              hipStream_t stream)
{
    const float* in_x     = (const float*)d_in[0];
    const float* in_pos   = (const float*)d_in[1];
    // d_in[2] = in_batch (implied by layout, unused)
    const float* skip_x   = (const float*)d_in[3];
    const float* skip_pos = (const float*)d_in[4];
    // d_in[5] = skip_batch (implied by layout, unused)
    const float* W1       = (const float*)d_in[6];
    const float* b1       = (const float*)d_in[7];
    const float* W2       = (const float*)d_in[8];
    const float* b2       = (const float*)d_in[9];
    // d_in[10] = knn_num == 3 (hardcoded)

    float* out  = (float*)d_out;
    float* Hcat = (float*)d_ws;                         // 16384*388 floats
    float* W1p  = Hcat + (size_t)NS_TOT * DPAD;         // 388*256 floats

    knn_interp_concat<<<NS_TOT / 256, 256, 0, stream>>>(in_x, in_pos, skip_x,
                                                        skip_pos, Hcat);
    pad_w1<<<DPAD, 256, 0, stream>>>(W1, W1p);

    const size_t ldsBytes = (size_t)(64 * DPAD + 64 * H1S) * sizeof(float); // 165,888
    fused_mlp<<<NS_TOT / 64, 256, ldsBytes, stream>>>(Hcat, W1p, b1, W2, b2, out);
}